// GNN_22960895165048
// MI455X (gfx1250) — compile-verified
//
#include <hip/hip_runtime.h>
#include <hip/hip_bf16.h>

// ---------------------------------------------------------------------------
// GNN 3-layer:  out_i = (sum_{e:dst=i} h_src - deg_i * h_i) @ W^T + deg_i * b
// Linearity: per-edge GEMM (78.6 GFLOP) -> per-node GEMM (4.9 GFLOP).
// Scatter is L2-resident on MI455X (h, agg = 25.6 MB each << 192 MB L2).
// GEMM: v_wmma_f32_16x16x4_f32; A/H tiles staged via
// GLOBAL_LOAD_ASYNC_TO_LDS_B128 (ASYNCcnt) with per-lane padded LDS addresses.
// ---------------------------------------------------------------------------

#define NNODES 50000
#define NEDGES 800000
#define DFEAT  128

typedef __attribute__((ext_vector_type(2))) float v2f;
typedef __attribute__((ext_vector_type(8))) float v8f;

#define WT_STRIDE 129           // padded row stride for W^T in LDS (dwords)
#define A_STRIDE  132           // padded row stride for agg/h tiles (dwords)
#define TILE_DW   (16 * A_STRIDE)            // one 16x128 padded tile
#define GEMM_WAVES 4
#define GEMM_SMEM_DW (DFEAT * WT_STRIDE + GEMM_WAVES * 2 * TILE_DW)
#define GEMM_SMEM_BYTES (GEMM_SMEM_DW * 4)   // 133,632 B (< 320 KB/WGP)

// ---------------------------------------------------------------------------
__global__ void gnn_zero_kernel(float* __restrict__ p, int n) {
    for (int i = blockIdx.x * blockDim.x + threadIdx.x; i < n;
         i += gridDim.x * blockDim.x)
        p[i] = 0.0f;
}

// deg[i] = number of edges whose dst == i (as float; reused by all 3 layers)
__global__ void gnn_deg_kernel(const int* __restrict__ ei,
                               float* __restrict__ deg, int nEdges) {
    int e = blockIdx.x * blockDim.x + threadIdx.x;
    if (e < nEdges) {
        int d = ei[nEdges + e];   // edge_index[1][e]
        __hip_atomic_fetch_add(&deg[d], 1.0f, __ATOMIC_RELAXED,
                               __HIP_MEMORY_SCOPE_AGENT);
    }
}

// agg[dst] += h[src]   — one wave32 per edge, float4 per lane (128 channels)
__global__ void gnn_scatter_kernel(const float* __restrict__ h,
                                   const int* __restrict__ ei,
                                   float* __restrict__ agg, int nEdges) {
    int g = blockIdx.x * blockDim.x + threadIdx.x;
    int e = g >> 5;
    int lane = g & 31;
    if (e >= nEdges) return;
    int s = ei[e];                // edge_index[0][e]  (source / x_j)
    int d = ei[nEdges + e];       // edge_index[1][e]  (target / x_i)
    const float4 v = ((const float4*)(h + (size_t)s * DFEAT))[lane];
    float* a = agg + (size_t)d * DFEAT + lane * 4;
    __hip_atomic_fetch_add(a + 0, v.x, __ATOMIC_RELAXED, __HIP_MEMORY_SCOPE_AGENT);
    __hip_atomic_fetch_add(a + 1, v.y, __ATOMIC_RELAXED, __HIP_MEMORY_SCOPE_AGENT);
    __hip_atomic_fetch_add(a + 2, v.z, __ATOMIC_RELAXED, __HIP_MEMORY_SCOPE_AGENT);
    __hip_atomic_fetch_add(a + 3, v.w, __ATOMIC_RELAXED, __HIP_MEMORY_SCOPE_AGENT);
}

// out[i,:] = (agg[i,:] - deg[i]*hin[i,:]) @ W^T + deg[i]*bias
// Block: 128 threads (4 waves), 64 rows per block (16 rows per wave).
// W^T staged in LDS (transposed, padded). agg/h tiles DMA'd raw into padded
// LDS via global_load_async_to_lds_b128; the deg-combine happens at fragment
// read (deg value is A-fragment-lane invariant).
__global__ void gnn_gemm_kernel(const float* __restrict__ agg,
                                const float* __restrict__ hin,
                                const float* __restrict__ deg,
                                const float* __restrict__ W,
                                const float* __restrict__ bias,
                                float* __restrict__ out, int nNodes) {
    extern __shared__ float smem[];
    float* Wt = smem;                               // [128][WT_STRIDE]
    const int tid  = threadIdx.x;
    const int wave = tid >> 5;
    const int lane = tid & 31;
    const int hi   = lane >> 4;        // 0 or 1 (lane half)
    const int ln   = lane & 15;

    float* AggW = smem + DFEAT * WT_STRIDE + wave * 2 * TILE_DW;  // raw agg tile
    float* HW   = AggW + TILE_DW;                                 // raw h tile

    // ---- stage W^T into LDS (coalesced float4 global loads, transpose) ----
    #pragma unroll
    for (int j = 0; j < (DFEAT * DFEAT / 4) / 128; ++j) {   // 32 iters
        int f4 = j * 128 + tid;                    // float4 index into W
        const float4 w4 = ((const float4*)W)[f4];
        int n = f4 >> 5;                           // row of W (output channel)
        int k = (f4 & 31) * 4;                     // col of W (input channel)
        Wt[(k + 0) * WT_STRIDE + n] = w4.x;
        Wt[(k + 1) * WT_STRIDE + n] = w4.y;
        Wt[(k + 2) * WT_STRIDE + n] = w4.z;
        Wt[(k + 3) * WT_STRIDE + n] = w4.w;
    }

    // ---- async DMA this wave's agg/h tiles into padded LDS (ASYNCcnt) ----
    const int rowbase = blockIdx.x * (GEMM_WAVES * 16) + wave * 16;
    {
        const unsigned ldsAgg = (unsigned)(uintptr_t)AggW;
        const unsigned ldsH   = (unsigned)(uintptr_t)HW;
        for (int r = 0; r < 16; ++r) {
            int row = rowbase + r;
            int rc  = row < nNodes ? row : 0;      // clamp; OOB rows never stored
            const float* gA = agg + (size_t)rc * DFEAT + lane * 4;
            const float* gH = hin + (size_t)rc * DFEAT + lane * 4;
            unsigned dA = ldsAgg + (unsigned)(r * A_STRIDE + lane * 4) * 4u;
            unsigned dH = ldsH   + (unsigned)(r * A_STRIDE + lane * 4) * 4u;
            asm volatile("global_load_async_to_lds_b128 %0, %1, off"
                         :: "v"(dA), "v"(gA) : "memory");
            asm volatile("global_load_async_to_lds_b128 %0, %1, off"
                         :: "v"(dH), "v"(gH) : "memory");
        }
    }
    // per-lane deg for the A fragment (lane L owns matrix row M = L%16)
    const int degRow = rowbase + ln;
    const float dvl = deg[degRow < nNodes ? degRow : 0];

    asm volatile("s_wait_asynccnt 0x0" ::: "memory");  // own-wave tiles ready
    __syncthreads();                                   // Wt ready (all waves)

    // ---- WMMA main loop: 8 N-tiles of 16, K in steps of 4 ----
    v8f acc[8];
    #pragma unroll
    for (int t = 0; t < 8; ++t) acc[t] = (v8f){0, 0, 0, 0, 0, 0, 0, 0};

    for (int k0 = 0; k0 < DFEAT; k0 += 4) {
        // A fragment: lane (hi,ln) holds A[M=ln][k0+2*hi .. +1],
        // A = agg - deg*h fused at read.
        const float2 ag = *(const float2*)(AggW + ln * A_STRIDE + k0 + 2 * hi);
        const float2 hh = *(const float2*)(HW   + ln * A_STRIDE + k0 + 2 * hi);
        v2f a;
        a[0] = ag.x - dvl * hh.x;
        a[1] = ag.y - dvl * hh.y;
        #pragma unroll
        for (int t = 0; t < 8; ++t) {
            // B fragment: lane holds B[k0+2*hi][n], B[k0+2*hi+1][n], n = t*16+ln
            v2f b;
            b[0] = Wt[(k0 + 2 * hi)     * WT_STRIDE + t * 16 + ln];
            b[1] = Wt[(k0 + 2 * hi + 1) * WT_STRIDE + t * 16 + ln];
            acc[t] = __builtin_amdgcn_wmma_f32_16x16x4_f32(
                false, a, false, b, (short)0, acc[t], false, false);
        }
    }

    // ---- epilogue: + deg[row]*bias[col], guarded stores ----
    #pragma unroll
    for (int t = 0; t < 8; ++t) {
        int col = t * 16 + ln;
        float bb = bias[col];
        #pragma unroll
        for (int r = 0; r < 8; ++r) {
            int row = rowbase + hi * 8 + r;        // C layout: M = r + 8*hi
            if (row < nNodes) {
                float dv = deg[row];
                out[(size_t)row * DFEAT + col] = acc[t][r] + dv * bb;
            }
        }
    }
}

// ---------------------------------------------------------------------------
extern "C" void kernel_launch(void* const* d_in, const int* in_sizes, int n_in,
                              void* d_out, int out_size, void* d_ws, size_t ws_size,
                              hipStream_t stream) {
    const float* x  = (const float*)d_in[0];
    const float* W1 = (const float*)d_in[1];
    const float* b1 = (const float*)d_in[2];
    const float* W2 = (const float*)d_in[3];
    const float* b2 = (const float*)d_in[4];
    const float* W3 = (const float*)d_in[5];
    const float* b3 = (const float*)d_in[6];
    const int*   ei = (const int*)d_in[7];   // edge_index [2][E]; d_in[8] unused
    float* out = (float*)d_out;

    const int N = in_sizes[0] / DFEAT;       // 50000
    const int E = in_sizes[7] / 2;           // 800000

    float* agg  = (float*)d_ws;              // N*DFEAT
    float* deg  = agg + (size_t)N * DFEAT;   // N
    float* hbuf = deg + N;                   // N*DFEAT

    const int zThreads = 256, zBlocks = 1024;
    const int eBlocks  = (E + 255) / 256;
    const int sBlocks  = (E * 32 + 255) / 256;           // one wave per edge
    const int gBlocks  = (N + GEMM_WAVES * 16 - 1) / (GEMM_WAVES * 16);

    // zero agg + deg (contiguous), then degree count (reused by all layers)
    gnn_zero_kernel<<<zBlocks, zThreads, 0, stream>>>(agg, N * DFEAT + N);
    gnn_deg_kernel<<<eBlocks, 256, 0, stream>>>(ei, deg, E);

    // ---- layer 1: x -> hbuf ----
    gnn_scatter_kernel<<<sBlocks, 256, 0, stream>>>(x, ei, agg, E);
    gnn_gemm_kernel<<<gBlocks, GEMM_WAVES * 32, GEMM_SMEM_BYTES, stream>>>(
        agg, x, deg, W1, b1, hbuf, N);
    gnn_zero_kernel<<<zBlocks, zThreads, 0, stream>>>(agg, N * DFEAT);

    // ---- layer 2: hbuf -> hbuf (in-place safe: blocks read only own rows) ----
    gnn_scatter_kernel<<<sBlocks, 256, 0, stream>>>(hbuf, ei, agg, E);
    gnn_gemm_kernel<<<gBlocks, GEMM_WAVES * 32, GEMM_SMEM_BYTES, stream>>>(
        agg, hbuf, deg, W2, b2, hbuf, N);
    gnn_zero_kernel<<<zBlocks, zThreads, 0, stream>>>(agg, N * DFEAT);

    // ---- layer 3: hbuf -> out ----
    gnn_scatter_kernel<<<sBlocks, 256, 0, stream>>>(hbuf, ei, agg, E);
    gnn_gemm_kernel<<<gBlocks, GEMM_WAVES * 32, GEMM_SMEM_BYTES, stream>>>(
        agg, hbuf, deg, W3, b3, out, N);
}